// QuantumKernelLayer_65481071405023
// MI455X (gfx1250) — compile-verified
//
#include <hip/hip_runtime.h>
#include <hip/hip_bf16.h>

typedef __attribute__((ext_vector_type(2))) float v2f;
typedef __attribute__((ext_vector_type(8))) float v8f;

#define NB 4096   // batch rows (x)
#define NS 4096   // support rows (sv)

// ---------------------------------------------------------------------------
// Phase 1: feature expansion.
// out[i][j] = prod_k cos((x_ik - s_jk)/2)
//           = prod_k ( cos(x_ik/2)cos(s_jk/2) + sin(x_ik/2)sin(s_jk/2) )
//           = sum_{m=0..15} A[i][m] * B[j][m]
// where A[i][m] = prod_k (bit_k(m) ? sin : cos)(x_ik/2), same for B from sv.
// One thread per row; writes 16 f32 features.
// ---------------------------------------------------------------------------
__global__ void qk_features(const float* __restrict__ x,
                            const float* __restrict__ sv,
                            float* __restrict__ A,
                            float* __restrict__ B) {
  int i = blockIdx.x * blockDim.x + threadIdx.x;   // 0 .. NB+NS-1
  const float* src;
  float* dst;
  int row;
  if (i < NB) { src = x;  dst = A; row = i; }
  else        { src = sv; dst = B; row = i - NB; }

  float c[4], s[4];
#pragma unroll
  for (int k = 0; k < 4; ++k) {
    float h = 0.5f * src[row * 4 + k];
    sincosf(h, &s[k], &c[k]);
  }
#pragma unroll
  for (int m = 0; m < 16; ++m) {
    float p = 1.0f;
#pragma unroll
    for (int k = 0; k < 4; ++k)
      p *= ((m >> k) & 1) ? s[k] : c[k];
    dst[row * 16 + m] = p;
  }
}

// ---------------------------------------------------------------------------
// Phase 2: out = |A (4096x16) * B^T (16x4096)| via V_WMMA_F32_16X16X4_F32.
// One wave per 16x16 output tile; K=16 done as 4 chained WMMAs (K=4 each).
//
// f32 16x16x4 operand layout (wave32):
//   A operand (16x4):  lane L holds row M=L%16; VGPR0=K0, VGPR1=K0+1 with
//                      K0 = 2*(L/16).
//   B operand (4x16):  symmetric — lane L holds col N=L%16, same K striping.
//   Since we multiply by B^T, both operands index their matrix as
//   [lane%16][k] -> identical load pattern for A and B.
//   C/D:  VGPR v holds row  v + 8*(lane/16), col lane%16.
// ---------------------------------------------------------------------------
__global__ void qk_wmma(const float* __restrict__ A,
                        const float* __restrict__ B,
                        float* __restrict__ out) {
  const int lane  = threadIdx.x & 31;
  const int wave  = threadIdx.x >> 5;
  const int tile  = blockIdx.x * (blockDim.x >> 5) + wave;  // 0 .. 256*256-1
  const int tm    = tile >> 8;    // output row tile (0..255)
  const int tn    = tile & 255;   // output col tile (0..255)

  const int r     = lane & 15;            // row of A / row of B (=col of out)
  const int kBase = (lane >> 4) << 1;     // 0 or 2

  const float* aRow = A + (size_t)(tm * 16 + r) * 16;
  const float* bRow = B + (size_t)(tn * 16 + r) * 16;

  v8f acc = {};
#pragma unroll
  for (int step = 0; step < 4; ++step) {
    const int k0 = step * 4 + kBase;            // even -> 8B aligned
    v2f a = *(const v2f*)(aRow + k0);
    v2f b = *(const v2f*)(bRow + k0);
    acc = __builtin_amdgcn_wmma_f32_16x16x4_f32(
        /*neg_a=*/false, a, /*neg_b=*/false, b,
        /*c_mod=*/(short)0, acc, /*reuse_a=*/false, /*reuse_b=*/false);
  }

  const int col     = tn * 16 + r;
  const int rowBase = tm * 16 + ((lane >> 4) << 3);  // +0 or +8
#pragma unroll
  for (int v = 0; v < 8; ++v) {
    out[(size_t)(rowBase + v) * NS + col] = fabsf(acc[v]);
  }
}

// ---------------------------------------------------------------------------
extern "C" void kernel_launch(void* const* d_in, const int* in_sizes, int n_in,
                              void* d_out, int out_size, void* d_ws, size_t ws_size,
                              hipStream_t stream) {
  const float* x  = (const float*)d_in[0];   // [4096, 4]
  const float* sv = (const float*)d_in[1];   // [4096, 4]
  float* outp = (float*)d_out;               // [4096, 4096]

  float* wsA = (float*)d_ws;                 // [4096, 16]  (256 KB)
  float* wsB = wsA + (size_t)NB * 16;        // [4096, 16]  (256 KB)

  // Phase 1: 8192 rows total, one thread each.
  {
    const int threads = 256;
    const int blocks  = (NB + NS) / threads;   // 32
    qk_features<<<blocks, threads, 0, stream>>>(x, sv, wsA, wsB);
  }

  // Phase 2: 256x256 = 65536 tiles, one wave (32 threads) per tile,
  // 8 waves per 256-thread block -> 8192 blocks.
  {
    const int threads = 256;
    const int wavesPerBlock = threads / 32;
    const int tiles = (NB / 16) * (NS / 16);
    const int blocks = tiles / wavesPerBlock;  // 8192
    qk_wmma<<<blocks, threads, 0, stream>>>(wsA, wsB, outp);
  }
}